// DiscriminativeLoss_41437844472370
// MI455X (gfx1250) — compile-verified
//
#include <hip/hip_runtime.h>

typedef float v2f __attribute__((ext_vector_type(2)));
typedef float v8f __attribute__((ext_vector_type(8)));

#define N_PIX 80000
#define B_ 16
#define L_ 8
#define D_ 16

// workspace layout (float offsets)
#define WS_SUMS  0      // 16*8*16 = 2048
#define WS_CNT   2048   // 16*8    = 128
#define WS_MEANS 2176   // 2048
#define WS_MSQ   4224   // 128
#define WS_PULL  4352   // 128
#define WS_TOTAL 4480

__global__ void zero_ws(float* __restrict__ ws) {
  const int i = blockIdx.x * blockDim.x + threadIdx.x;
  if (i < WS_TOTAL) ws[i] = 0.f;
}

// Pass 1: sums[b][l][d] = sum_n mask[l][n] * emb[d][n]  via WMMA f32 16x16x4,
// counts[b][l] = sum_n mask[l][n].
// grid = (25, 16); 3200 px per block; each wave: 50 steps x 8 px (b128 loads,
// peeled double-buffer -> branch-free inner loop), 2 WMMAs per step.
// WMMA K-mapping: K={0,1} <- first two pixels of lane-half-0's quad,
// K={2,3} <- lane-half-1's quad (consistent between A and B).
__global__ __launch_bounds__(256) void k1_sums(const float* __restrict__ emb,
                                               const int* __restrict__ seg,
                                               float* __restrict__ ws) {
  __shared__ float s_sums[L_][D_];
  __shared__ float s_cnt[L_];
  const int tid = threadIdx.x;
  const int b = blockIdx.y;
  if (tid < L_ * D_) s_sums[tid >> 4][tid & 15] = 0.f;
  if (tid < L_) s_cnt[tid] = 0.f;
  __syncthreads();

  const int lane  = tid & 31;
  const int wave  = tid >> 5;
  const int m     = lane & 15;            // A row (l) / B col (d)
  const int khalf = lane >> 4;            // which K-half this lane supplies
  const float rv  = (m < L_) ? 1.f : 0.f; // rows 8..15 of A are zero
  const int nbase = blockIdx.x * 3200;

  const int*   segrow = seg + (size_t)b * L_ * N_PIX + (size_t)(m & 7) * N_PIX;
  const float* embrow = emb + (size_t)b * D_ * N_PIX + (size_t)m * N_PIX;

  v8f c = {0.f, 0.f, 0.f, 0.f, 0.f, 0.f, 0.f, 0.f};
  float cnt = 0.f;

  auto step = [&](const int4 mc, const float4 ec) {
    const float a0 = mc.x > 0 ? rv : 0.f;
    const float a1 = mc.y > 0 ? rv : 0.f;
    const float a2 = mc.z > 0 ? rv : 0.f;
    const float a3 = mc.w > 0 ? rv : 0.f;
    cnt += (a0 + a1) + (a2 + a3);
    v2f a, bm;
    a.x = a0;   a.y = a1;   bm.x = ec.x; bm.y = ec.y;
    c = __builtin_amdgcn_wmma_f32_16x16x4_f32(false, a, false, bm, (short)0, c, false, false);
    a.x = a2;   a.y = a3;   bm.x = ec.z; bm.y = ec.w;
    c = __builtin_amdgcn_wmma_f32_16x16x4_f32(false, a, false, bm, (short)0, c, false, false);
  };

  // peeled double-buffer: 49 steps with unconditional prefetch + 1 epilogue step
  int k0 = nbase + wave * 8 + khalf * 4;
  int4   mi = *(const int4*)(segrow + k0);
  float4 ev = *(const float4*)(embrow + k0);
#pragma unroll 1
  for (int it = 0; it < 49; ++it) {
    const int4   mc = mi;
    const float4 ec = ev;
    k0 += 64;                              // 8 waves * 8 px
    mi = *(const int4*)(segrow + k0);
    ev = *(const float4*)(embrow + k0);
    step(mc, ec);
  }
  step(mi, ev);

  cnt += __shfl_xor(cnt, 16, 32);          // fold K halves: lane l + lane l+16
  if (lane < L_) atomicAdd(&s_cnt[lane], cnt);
  if (lane < 16) {                         // C rows 0..7 live in VGPR 0..7 of lanes 0..15
#pragma unroll
    for (int r = 0; r < 8; ++r) atomicAdd(&s_sums[r][lane], c[r]);
  }
  __syncthreads();
  if (tid < L_ * D_) atomicAdd(&ws[WS_SUMS + b * (L_ * D_) + tid], s_sums[tid >> 4][tid & 15]);
  if (tid < L_)      atomicAdd(&ws[WS_CNT + b * L_ + tid], s_cnt[tid]);
}

// means[b][l][d] = sums / max(cnt,1);  msq[b][l] = sum_d means^2
__global__ __launch_bounds__(256) void k2_means(float* __restrict__ ws) {
  const int tid = threadIdx.x;
  for (int i = tid; i < B_ * L_ * D_; i += 256) {
    const float cnt = ws[WS_CNT + (i >> 4)];
    ws[WS_MEANS + i] = ws[WS_SUMS + i] / fmaxf(cnt, 1.f);
  }
  __syncthreads();
  for (int i = tid; i < B_ * L_; i += 256) {
    float s = 0.f;
    for (int d = 0; d < D_; ++d) {
      const float mv = ws[WS_MEANS + i * D_ + d];
      s += mv * mv;
    }
    ws[WS_MSQ + i] = s;
  }
}

// Pass 2: pull[b][l] = sum_n mask * relu(||e_n - mean_l|| - 0.5)^2
// cross[l][j] via 4x WMMA f32 16x16x4 (K = d chunks).  grid = (25, 16): 3200 px/block.
// Each wave: 25 tiles of 16 px, peeled double-buffer over (8 emb + 8 mask) loads.
__global__ __launch_bounds__(256) void k3_pull(const float* __restrict__ emb,
                                               const int* __restrict__ seg,
                                               float* __restrict__ ws) {
  __shared__ float s_pull[L_];
  __shared__ float s_msq[L_];
  const int tid = threadIdx.x;
  const int b = blockIdx.y;
  if (tid < L_) { s_pull[tid] = 0.f; s_msq[tid] = ws[WS_MSQ + b * L_ + tid]; }
  __syncthreads();

  const int lane = tid & 31;
  const int wave = tid >> 5;
  const int m    = lane & 15;  // A row (l) / B col (pixel j)
  const int hi   = lane >> 4;
  const float lanev = (lane < 16) ? 1.f : 0.f;  // C rows 8..15 (upper half-wave) are zero

  // A tiles: means, 4 chunks of K=4 along d (constant over the pixel loop)
  v2f a[4];
#pragma unroll
  for (int kc = 0; kc < 4; ++kc) {
    float ax = 0.f, ay = 0.f;
    if (m < L_) {
      const int base = WS_MEANS + b * (L_ * D_) + m * D_ + kc * 4 + hi * 2;
      ax = ws[base]; ay = ws[base + 1];
    }
    a[kc].x = ax; a[kc].y = ay;
  }

  const float* embb = emb + (size_t)b * D_ * N_PIX + (size_t)(hi * 2) * N_PIX + m;
  const int*   segb = seg + (size_t)b * L_ * N_PIX + m;
  const int nbase = blockIdx.x * 3200;

  float acc[8] = {0.f, 0.f, 0.f, 0.f, 0.f, 0.f, 0.f, 0.f};
  float eb[8];
  int   mb[8];

  auto loadTile = [&](int n0) {
#pragma unroll
    for (int kc = 0; kc < 4; ++kc) {
      const float* p = embb + (size_t)(kc * 4) * N_PIX + n0;
      eb[kc * 2]     = p[0];
      eb[kc * 2 + 1] = p[N_PIX];
    }
#pragma unroll
    for (int r = 0; r < 8; ++r) mb[r] = segb[(size_t)r * N_PIX + n0];
  };

  auto compute = [&](const float ec[8], const int mc[8]) {
    v8f c = {0.f, 0.f, 0.f, 0.f, 0.f, 0.f, 0.f, 0.f};
    float esq = 0.f;
#pragma unroll
    for (int kc = 0; kc < 4; ++kc) {
      v2f bm; bm.x = ec[kc * 2]; bm.y = ec[kc * 2 + 1];   // B: emb 4(d) x 16(px)
      esq += bm.x * bm.x + bm.y * bm.y;
      c = __builtin_amdgcn_wmma_f32_16x16x4_f32(false, a[kc], false, bm, (short)0, c, false, false);
    }
    esq += __shfl_xor(esq, 16, 32);           // full sum_d e^2 in lanes 0..15
#pragma unroll
    for (int r = 0; r < 8; ++r) {
      const float mk = (mc[r] > 0) ? lanev : 0.f;
      float d2 = fmaxf(esq - 2.f * c[r] + s_msq[r], 0.f);
      const float dist = __builtin_sqrtf(d2);
      const float h = fmaxf(dist - 0.5f, 0.f);
      acc[r] += mk * h * h;                   // mk == 0 for upper half-wave
    }
  };

  // peeled double-buffer: 24 tiles with unconditional prefetch + 1 epilogue tile
  int n0 = nbase + wave * 16;
  loadTile(n0);
#pragma unroll 1
  for (int it = 0; it < 24; ++it) {
    float ec[8]; int mc[8];
#pragma unroll
    for (int i = 0; i < 8; ++i) { ec[i] = eb[i]; mc[i] = mb[i]; }
    n0 += 128;                                // 8 waves * 16 px
    loadTile(n0);
    compute(ec, mc);
  }
  compute(eb, mb);

  if (lane < 16) {
#pragma unroll
    for (int r = 0; r < 8; ++r) atomicAdd(&s_pull[r], acc[r]);
  }
  __syncthreads();
  if (tid < L_) atomicAdd(&ws[WS_PULL + b * L_ + tid], s_pull[tid]);
}

// Finalize: var loss + pairwise push loss (trivial work).
__global__ __launch_bounds__(128) void k4_final(const float* __restrict__ ws,
                                                float* __restrict__ out) {
  __shared__ float s_var[128];
  __shared__ float s_dist[16];
  const int tid = threadIdx.x;

  float v = 0.f;
  {
    const float cnt = ws[WS_CNT + tid];
    if (cnt > 0.f) v = ws[WS_PULL + tid] / cnt;
  }
  s_var[tid] = v;

  if (tid < 16) {
    const int b = tid;
    float valid[8];
    float nv = 0.f;
    for (int l = 0; l < 8; ++l) {
      valid[l] = ws[WS_CNT + b * 8 + l] > 0.f ? 1.f : 0.f;
      nv += valid[l];
    }
    float hsum = 0.f;
    for (int l1 = 0; l1 < 8; ++l1)
      for (int l2 = 0; l2 < 8; ++l2) {
        if (l1 == l2) continue;   // diagonal hinge is exactly 0 in the reference
        float d2 = 0.f;
        for (int d = 0; d < 16; ++d) {
          const float diff = ws[WS_MEANS + b * 128 + l1 * 16 + d] -
                             ws[WS_MEANS + b * 128 + l2 * 16 + d];
          d2 += diff * diff;
        }
        const float dist = __builtin_sqrtf(d2);
        const float h = fmaxf(3.0f - dist, 0.f);
        hsum += h * h * valid[l1] * valid[l2];
      }
    float per_b = 0.f;
    if (nv > 1.f) per_b = hsum / fmaxf(nv * (nv - 1.f), 1.f) * 0.5f;
    s_dist[b] = per_b;
  }
  __syncthreads();
  if (tid == 0) {
    float vs = 0.f;
    for (int i = 0; i < 128; ++i) vs += s_var[i];
    float ds = 0.f;
    for (int i = 0; i < 16; ++i) ds += s_dist[i];
    out[0] = vs / (float)(B_ * L_);  // /L/B
    out[1] = ds / (float)B_;
    out[2] = 0.f;
  }
}

extern "C" void kernel_launch(void* const* d_in, const int* in_sizes, int n_in,
                              void* d_out, int out_size, void* d_ws, size_t ws_size,
                              hipStream_t stream) {
  const float* emb = (const float*)d_in[0];
  const int*   seg = (const int*)d_in[1];
  float* ws  = (float*)d_ws;
  float* out = (float*)d_out;

  zero_ws<<<(WS_TOTAL + 255) / 256, 256, 0, stream>>>(ws);
  k1_sums<<<dim3(25, 16), 256, 0, stream>>>(emb, seg, ws);   // 80000 = 25 * 3200
  k2_means<<<1, 256, 0, stream>>>(ws);
  k3_pull<<<dim3(25, 16), 256, 0, stream>>>(emb, seg, ws);   // 80000 = 25 * 3200
  k4_final<<<1, 128, 0, stream>>>(ws, out);
}